// Correlation1D_60911226192132
// MI455X (gfx1250) — compile-verified
//
#include <hip/hip_runtime.h>

// Correlation1D (FlowNet cost volume) on gfx1250 via V_WMMA_F32_16X16X4_F32.
//
// out[b,d,h,w] = (1/C) * sum_c in1[b,c,h,w] * in2[b,c,h,w+d-40], d in [0,81)
//
// Per (b,h): banded Gram matrix of two 192x256 position-major matrices.
// One wave32 owns TWO adjacent 16-wide w-tiles (w0, w0+16): their 7-tile
// diagonal bands share 6 v-tiles, so one load of 8 B-tiles feeds 14 WMMAs
// per K-step (20 b32 loads / 14 wmma = 1.43, vs 2.29 for the 1-tile version),
// and cuts in2 L2 traffic ~40%. fp32 WMMA keeps full precision: at ~17
// FLOP/byte this problem sits near the f32-matrix roofline crossover
// (HBM floor ~15us vs ~10us of f32 WMMA math), so bf16 gains nothing.

typedef float v2f __attribute__((ext_vector_type(2)));
typedef float v8f __attribute__((ext_vector_type(8)));

#define C_CH      256
#define H_DIM     96
#define W_DIM     192
#define D_OUT     81
#define CH_STRIDE (H_DIM * W_DIM)   // 18432 elements between channels
#define NT        7                 // v-tiles per w-tile
#define NTB       8                 // union of v-tiles for a w-tile pair
#define LDS_PITCH 17                // 16 + 1 pad -> conflict-free LDS

// K-loop specialized on wave-uniform MASKED: interior pairs (B span fully in
// [0,192)) skip the 16 mask multiplies per step.
template <bool MASKED>
__device__ __forceinline__ void corr_kloop(const float* __restrict__ p1,
                                           const float* __restrict__ p2,
                                           int aoff0, int aoff1, int ksub,
                                           const int (&boff)[NTB],
                                           const float (&bmsk)[NTB],
                                           v8f (&acc)[2][NT]) {
#pragma unroll 2
  for (int k = 0; k < C_CH; k += 4) {
    const size_t c0 = (size_t)(k + ksub) * CH_STRIDE;  // this half's channels
    const size_t c1 = c0 + CH_STRIDE;
    // fp32 A/B VGPR layout (ISA 7.12.2): lane l<16 -> row li, VGPR0/1 = K+0/1;
    // lanes 16..31 -> K+2/3. Position-major for both operands of a Gram product.
    v2f a0, a1;
    a0.x = p1[c0 + aoff0];  a0.y = p1[c1 + aoff0];
    a1.x = p1[c0 + aoff1];  a1.y = p1[c1 + aoff1];
    v2f bb[NTB];
#pragma unroll
    for (int j = 0; j < NTB; ++j) {
      float x = p2[c0 + boff[j]];
      float y = p2[c1 + boff[j]];
      if (MASKED) { x *= bmsk[j]; y *= bmsk[j]; }  // zero-pad outside W
      bb[j].x = x;  bb[j].y = y;
    }
#pragma unroll
    for (int t = 0; t < NT; ++t) {
      // D = A(16x4)*B(4x16)+C ; args: neg_a, A, neg_b, B, c_mod, C, reuse_a, reuse_b
      acc[0][t] = __builtin_amdgcn_wmma_f32_16x16x4_f32(
          false, a0, false, bb[t], (short)0, acc[0][t], false, false);
      acc[1][t] = __builtin_amdgcn_wmma_f32_16x16x4_f32(
          false, a1, false, bb[t + 1], (short)0, acc[1][t], false, false);
    }
  }
}

__global__ __launch_bounds__(32)
void corr1d_wmma_f32_kernel(const float* __restrict__ in1,
                            const float* __restrict__ in2,
                            float* __restrict__ out) {
  const int l  = threadIdx.x;   // 0..31
  const int hl = l >> 4;        // half-wave: 0 or 1
  const int li = l & 15;

  int blk = blockIdx.x;         // (b*H + h)*6 + pair
  const int pr = blk % 6;       blk /= 6;
  const int h  = blk % H_DIM;
  const int b  = blk / H_DIM;
  const int w0 = pr * 32;       // this wave covers w0..w0+31

  const size_t base = (size_t)b * C_CH * CH_STRIDE + (size_t)h * W_DIM;
  const float* __restrict__ p1 = in1 + base;
  const float* __restrict__ p2 = in2 + base;

  // 8 B-tiles at columns w0-48+16j, j=0..7; tile pair 0 uses j=0..6, pair 1
  // uses j=1..7. Clamped per-lane column + zero-mask (hoisted out of K loop).
  int   boff[NTB];
  float bmsk[NTB];
#pragma unroll
  for (int j = 0; j < NTB; ++j) {
    int v = w0 - 48 + 16 * j + li;
    bool ok = (v >= 0) && (v < W_DIM);
    boff[j] = ok ? v : 0;
    bmsk[j] = ok ? 1.0f : 0.0f;
  }
  const int aoff0 = w0 + li;
  const int aoff1 = w0 + 16 + li;
  const int ksub  = 2 * hl;

  v8f acc[2][NT];
#pragma unroll
  for (int tt = 0; tt < 2; ++tt)
#pragma unroll
    for (int t = 0; t < NT; ++t)
      acc[tt][t] = (v8f){0.f, 0.f, 0.f, 0.f, 0.f, 0.f, 0.f, 0.f};

  const bool needs_mask = (w0 < 48) || (w0 + 80 > W_DIM);  // wave-uniform
  if (needs_mask)
    corr_kloop<true >(p1, p2, aoff0, aoff1, ksub, boff, bmsk, acc);
  else
    corr_kloop<false>(p1, p2, aoff0, aoff1, ksub, boff, bmsk, acc);

  // Band extraction: acc[tt][t] element (VGPR r, lane l) is
  // G[w0+16tt+m, w0+16tt-48+16t+n], m=r+8*hl, n=li -> d' = d+8 = 16t+n-m.
  // Every d' in [0,96) hit exactly once per (tile, m). Transpose via LDS so
  // global stores are full-64B-line coalesced along w.
  __shared__ float lds[2 * 96 * LDS_PITCH];
  const float scale = 1.0f / (float)C_CH;
#pragma unroll
  for (int tt = 0; tt < 2; ++tt) {
#pragma unroll
    for (int t = 0; t < NT; ++t) {
#pragma unroll
      for (int r = 0; r < 8; ++r) {
        int m  = r + 8 * hl;
        int dp = 16 * t + li - m;          // [-15,111]; keep [0,96)
        if (dp >= 0 && dp < 96)
          lds[(tt * 96 + dp) * LDS_PITCH + m] = acc[tt][t][r] * scale;
      }
    }
  }
  __syncthreads();

  // 81 displacement rows x 2 tiles; lanes 0-15 -> row 2i, lanes 16-31 -> row
  // 2i+1: each b32 store writes two full 64B lines.
#pragma unroll
  for (int tt = 0; tt < 2; ++tt) {
    float* __restrict__ op =
        out + ((size_t)b * D_OUT * H_DIM + (size_t)h) * W_DIM + w0 + 16 * tt + li;
#pragma unroll
    for (int i = 0; i < 41; ++i) {
      int row = 2 * i + hl;
      if (row < D_OUT)
        op[(size_t)row * CH_STRIDE] = lds[(tt * 96 + row + 8) * LDS_PITCH + li];
    }
  }
}

extern "C" void kernel_launch(void* const* d_in, const int* in_sizes, int n_in,
                              void* d_out, int out_size, void* d_ws, size_t ws_size,
                              hipStream_t stream) {
  (void)in_sizes; (void)n_in; (void)out_size; (void)d_ws; (void)ws_size;
  const float* in1 = (const float*)d_in[0];
  const float* in2 = (const float*)d_in[1];
  float* out = (float*)d_out;

  const int B = 8;
  dim3 grid(B * H_DIM * (W_DIM / 32));  // 8 * 96 * 6 = 4608 single-wave WGs
  dim3 block(32);
  corr1d_wmma_f32_kernel<<<grid, block, 0, stream>>>(in1, in2, out);
}